// CapsuleLayer_87153476370889
// MI455X (gfx1250) — compile-verified
//
#include <hip/hip_runtime.h>
#include <math.h>

typedef __attribute__((ext_vector_type(2))) float v2f;
typedef __attribute__((ext_vector_type(4))) float v4f;
typedef __attribute__((ext_vector_type(8))) float v8f;

#define BATCH 64
#define IC    2048
#define ID    8
#define OC    32
#define OD    16
#define EPSQ  1e-9f

// ---------------------------------------------------------------------------
// Kernel 1: u_hat[b][o][i][j] = sum_d W[i][o][j][d] * x[b][i][d]
// Per i: GEMM [16b x 8d] x [8d x 16j] per (o, btile) using V_WMMA_F32_16X16X4_F32
// grid = (IC, 4 btiles), 256 thr = 8 waves, wave w covers o = 4w..4w+3.
// A frag (16x4 f32): lanes 0-15 -> {K=0,K=1}, lanes 16-31 -> {K=2,K=3}
// B frag (4x16 f32): same split on K, N = lane&15
// D frag: VGPR r -> M = r + 8*(lane>>4), N = lane&15
// ---------------------------------------------------------------------------
__global__ __launch_bounds__(256) void uhat_wmma(const float* __restrict__ x,
                                                 const float* __restrict__ W,
                                                 float* __restrict__ u) {
  const int i     = blockIdx.x;
  const int btile = blockIdx.y;
  const int wave  = threadIdx.x >> 5;
  const int lane  = threadIdx.x & 31;
  const int row   = lane & 15;   // M for A, N (=j) for B/D
  const int hi    = lane >> 4;   // K-half selector

  // A: x[btile*16+row, i, 0..7]
  const float* xp = x + ((size_t)(btile * 16 + row) * IC + i) * ID;
  v2f a0 = *(const v2f*)(xp + hi * 2);        // K = 0..3 slice
  v2f a1 = *(const v2f*)(xp + 4 + hi * 2);    // K = 4..7 slice

  for (int oo = 0; oo < 4; ++oo) {
    const int o = wave * 4 + oo;
    const float* wp = W + (((size_t)i * OC + o) * OD + row) * ID;
    v2f b0 = *(const v2f*)(wp + hi * 2);
    v2f b1 = *(const v2f*)(wp + 4 + hi * 2);

    v8f acc = {0.f, 0.f, 0.f, 0.f, 0.f, 0.f, 0.f, 0.f};
    acc = __builtin_amdgcn_wmma_f32_16x16x4_f32(false, a0, false, b0,
                                                (short)0, acc, false, false);
    acc = __builtin_amdgcn_wmma_f32_16x16x4_f32(false, a1, false, b1,
                                                (short)0, acc, false, false);

#pragma unroll
    for (int r = 0; r < 8; ++r) {
      const int bidx = btile * 16 + r + hi * 8;   // batch index (M)
      u[(((size_t)bidx * OC + o) * IC + i) * OD + row] = acc[r];
    }
  }
}

// ---------------------------------------------------------------------------
// Kernel 2: c[b][i][o] = softmax_o(bl[b][i][o]); one wave per (b,i) row, lane=o
// ---------------------------------------------------------------------------
__global__ __launch_bounds__(256) void softmax_c(const float* __restrict__ bl,
                                                 float* __restrict__ c) {
  const int rowid = blockIdx.x * 8 + (threadIdx.x >> 5);  // flat b*IC + i
  const int o     = threadIdx.x & 31;
  const size_t idx = (size_t)rowid * OC + o;
  float v = bl[idx];
  float m = v;
#pragma unroll
  for (int k = 16; k >= 1; k >>= 1) m = fmaxf(m, __shfl_xor(m, k, 32));
  float e = __expf(v - m);
  float s = e;
#pragma unroll
  for (int k = 16; k >= 1; k >>= 1) s += __shfl_xor(s, k, 32);
  c[idx] = e / s;
}

// ---------------------------------------------------------------------------
// Kernel 3: s_j[b][o][j] = sum_i c[b][i][o]*u[b][o][i][j]; then squash -> vout
// block per (b,o); thread = (ig 0..15, j 0..15); streams contiguous 128KB plane
// ---------------------------------------------------------------------------
__global__ __launch_bounds__(256) void sj_squash(const float* __restrict__ u,
                                                 const float* __restrict__ c,
                                                 float* __restrict__ vout) {
  const int o  = blockIdx.x;
  const int b  = blockIdx.y;
  const int j  = threadIdx.x & 15;
  const int ig = threadIdx.x >> 4;
  __shared__ float red[256];
  __shared__ float s2sh;

  const float* up = u + ((size_t)b * OC + o) * IC * OD;
  const float* cp = c + (size_t)b * IC * OC + o;

  float acc = 0.f;
  for (int i = ig; i < IC; i += 16)
    acc += cp[(size_t)i * OC] * up[(size_t)i * OD + j];

  red[ig * 16 + j] = acc;
  __syncthreads();
#pragma unroll
  for (int s = 8; s >= 1; s >>= 1) {
    if (ig < s) red[ig * 16 + j] += red[(ig + s) * 16 + j];
    __syncthreads();
  }
  if (threadIdx.x == 0) {
    float s2 = 0.f;
#pragma unroll
    for (int jj = 0; jj < OD; ++jj) { const float t = red[jj]; s2 += t * t; }
    s2sh = s2;
  }
  __syncthreads();
  if (threadIdx.x < OD) {
    const float s  = red[j];
    const float s2 = s2sh;
    const float scale = (s2 / (1.f + s2)) * rsqrtf(s2 + EPSQ);
    vout[((size_t)b * OC + o) * OD + j] = s * scale;
  }
}

// ---------------------------------------------------------------------------
// Kernel 4: bl[b][i][o] += sum_j u[b][o][i][j] * v[b][o][j]
// block per (b, i-chunk of 8); thread = (il 0..7, o 0..31); v staged in LDS
// ---------------------------------------------------------------------------
__global__ __launch_bounds__(256) void b_update(const float* __restrict__ u,
                                                const float* __restrict__ v,
                                                float* __restrict__ bl) {
  const int b  = blockIdx.y;
  const int o  = threadIdx.x & 31;
  const int il = threadIdx.x >> 5;
  const int i  = blockIdx.x * 8 + il;
  __shared__ float vsh[OC * OD];
  for (int t = threadIdx.x; t < OC * OD; t += 256)
    vsh[t] = v[(size_t)b * OC * OD + t];
  __syncthreads();

  const v4f* u4 = (const v4f*)(u + (((size_t)b * OC + o) * IC + i) * OD);
  const float* vp = &vsh[o * OD];
  float dot = 0.f;
#pragma unroll
  for (int q = 0; q < 4; ++q) {
    const v4f uu = u4[q];
    dot += uu.x * vp[q * 4 + 0] + uu.y * vp[q * 4 + 1] +
           uu.z * vp[q * 4 + 2] + uu.w * vp[q * 4 + 3];
  }
  bl[((size_t)b * IC + i) * OC + o] += dot;
}

// ---------------------------------------------------------------------------
extern "C" void kernel_launch(void* const* d_in, const int* in_sizes, int n_in,
                              void* d_out, int out_size, void* d_ws, size_t ws_size,
                              hipStream_t stream) {
  (void)in_sizes; (void)n_in; (void)out_size; (void)ws_size;
  const float* x = (const float*)d_in[0];  // [64,2048,8]
  const float* W = (const float*)d_in[1];  // [1,2048,32,16,8]

  char* ws = (char*)d_ws;
  const size_t u_bytes  = (size_t)BATCH * OC * IC * OD * sizeof(float); // 256 MiB
  const size_t bc_bytes = (size_t)BATCH * IC * OC * sizeof(float);      // 16 MiB
  float* u  = (float*)ws;
  float* bl = (float*)(ws + u_bytes);
  float* c  = (float*)(ws + u_bytes + bc_bytes);
  float* v  = (float*)(ws + u_bytes + 2 * bc_bytes);

  hipMemsetAsync(bl, 0, bc_bytes, stream);
  uhat_wmma<<<dim3(IC, 4), 256, 0, stream>>>(x, W, u);

  for (int r = 0; r < 3; ++r) {
    softmax_c<<<dim3(BATCH * IC / 8), 256, 0, stream>>>(bl, c);
    float* vt = (r == 2) ? (float*)d_out : v;
    sj_squash<<<dim3(OC, BATCH), 256, 0, stream>>>(u, c, vt);
    if (r < 2)
      b_update<<<dim3(IC / 8, BATCH), 256, 0, stream>>>(u, v, bl);
  }
}